// axis_aligned_gmm_layer_57174604645137
// MI455X (gfx1250) — compile-verified
//
#include <hip/hip_runtime.h>
#include <hip/hip_bf16.h>
#include <math.h>

// Shapes fixed by the reference.
#define BATCH 4096
#define KCLS  256
#define DIM   256
#define AD    512   // augmented inner dim: [x^2 | x]

typedef __attribute__((ext_vector_type(2))) float v2f;
typedef __attribute__((ext_vector_type(8))) float v8f;

// ---------------------------------------------------------------------------
// Phase 1: per-class preprocessing.
//   W[k][0:256]   = Dinv
//   W[k][256:512] = -2 * c * Dinv
//   bias[k]       = -0.5 * ( sum_d c^2*Dinv + sum_d log(D) )
// One block per class, 256 threads (one per dim).
// ---------------------------------------------------------------------------
__global__ __launch_bounds__(256) void gmm_prep_kernel(
    const float* __restrict__ centers, const float* __restrict__ Droot,
    float* __restrict__ W, float* __restrict__ bias) {
  const int k = blockIdx.x;
  const int d = threadIdx.x;

  const float dr   = Droot[k * DIM + d];
  const float Dv   = fabsf(dr) + 1e-4f;
  const float Dinv = 1.0f / Dv;
  const float c    = centers[k * DIM + d];

  W[k * AD + d]       = Dinv;
  W[k * AD + DIM + d] = -2.0f * c * Dinv;

  __shared__ float s1[256];
  __shared__ float s2[256];
  s1[d] = c * c * Dinv;
  s2[d] = logf(Dv);
  __syncthreads();
#pragma unroll
  for (int off = 128; off > 0; off >>= 1) {
    if (d < off) { s1[d] += s1[d + off]; s2[d] += s2[d + off]; }
    __syncthreads();
  }
  if (d == 0) bias[k] = -0.5f * (s1[0] + s2[0]);
}

// ---------------------------------------------------------------------------
// Phase 2: fused GEMM (f32 WMMA 16x16x4) + row softmax.
// One workgroup = 8 waves = 16 batch rows x all 256 classes.
//   score[b,k] = -0.5 * sum_d A[b,d]*W[k,d] + bias[k];  resp = softmax_k(score)
// ---------------------------------------------------------------------------
__global__ __launch_bounds__(256) void gmm_main_kernel(
    const float* __restrict__ x, const float* __restrict__ W,
    const float* __restrict__ bias, float* __restrict__ out) {
  constexpr int AS = AD + 4;    // A-tile LDS row stride (floats): bank-conflict-free b64 frag loads
  constexpr int SS = KCLS + 4;  // score LDS row stride

  __shared__ float ldsA[16 * AS];
  __shared__ float ldsS[16 * SS];
  __shared__ float rowMax[16];
  __shared__ float rowInv[16];

  const int tid     = threadIdx.x;
  const int rowbase = blockIdx.x * 16;

  // ---- Stage A tile into LDS: row r = [x^2 (256) | x (256)], coalesced float4 loads
#pragma unroll
  for (int i = 0; i < 4; ++i) {
    const int lin = tid + 256 * i;       // 0..1023  (16 rows x 64 float4)
    const int r   = lin >> 6;
    const int c4  = (lin & 63) << 2;
    const float4 v = *reinterpret_cast<const float4*>(&x[(rowbase + r) * DIM + c4]);
    float* arow = &ldsA[r * AS];
    arow[c4 + 0] = v.x * v.x;  arow[c4 + 1] = v.y * v.y;
    arow[c4 + 2] = v.z * v.z;  arow[c4 + 3] = v.w * v.w;
    arow[DIM + c4 + 0] = v.x;  arow[DIM + c4 + 1] = v.y;
    arow[DIM + c4 + 2] = v.z;  arow[DIM + c4 + 3] = v.w;
  }
  __syncthreads();

  // ---- WMMA fragment coordinates (wave32 layouts per CDNA5 ISA 7.12.2)
  const int wave = tid >> 5;
  const int lane = tid & 31;
  const int lrow = lane & 15;   // A: M index / B: N index
  const int lhi  = lane >> 4;   // K-half selector
  const int cb0  = wave * 32 + lrow;   // class owned by this lane, col tile 0
  const int cb1  = cb0 + 16;           // col tile 1

  v8f acc0 = {};
  v8f acc1 = {};
  const float* aPtr  = &ldsA[lrow * AS + 2 * lhi];
  const float* b0Ptr = &W[cb0 * AD + 2 * lhi];
  const float* b1Ptr = &W[cb1 * AD + 2 * lhi];

#pragma unroll 4
  for (int d = 0; d < AD; d += 4) {
    const v2f a  = *reinterpret_cast<const v2f*>(aPtr + d);    // ds_load_b64
    const v2f b0 = *reinterpret_cast<const v2f*>(b0Ptr + d);   // global_load_b64 (L2-resident W)
    const v2f b1 = *reinterpret_cast<const v2f*>(b1Ptr + d);
    acc0 = __builtin_amdgcn_wmma_f32_16x16x4_f32(false, a, false, b0,
                                                 (short)0, acc0, false, false);
    acc1 = __builtin_amdgcn_wmma_f32_16x16x4_f32(false, a, false, b1,
                                                 (short)0, acc1, false, false);
  }

  // ---- score = -0.5*acc + bias[class]; scatter C tiles into LDS score buffer
  const float bv0 = bias[cb0];
  const float bv1 = bias[cb1];
#pragma unroll
  for (int j = 0; j < 8; ++j) {
    const int m = j + 8 * lhi;  // C layout: vgpr j, lane half -> row
    ldsS[m * SS + cb0] = -0.5f * acc0[j] + bv0;
    ldsS[m * SS + cb1] = -0.5f * acc1[j] + bv1;
  }
  __syncthreads();

  // ---- Row softmax over 256 classes. 16 threads per row, 16 cols each.
  const int r   = tid >> 4;
  const int c16 = tid & 15;
  const float* srow = &ldsS[r * SS];

  float mx = -__builtin_inff();
#pragma unroll
  for (int j = 0; j < 16; ++j) mx = fmaxf(mx, srow[c16 + 16 * j]);
#pragma unroll
  for (int m = 1; m < 16; m <<= 1) mx = fmaxf(mx, __shfl_xor(mx, m, 16));

  float sum = 0.0f;
#pragma unroll
  for (int j = 0; j < 16; ++j) sum += expf(srow[c16 + 16 * j] - mx);
#pragma unroll
  for (int m = 1; m < 16; m <<= 1) sum += __shfl_xor(sum, m, 16);

  if (c16 == 0) { rowMax[r] = mx; rowInv[r] = 1.0f / sum; }
  __syncthreads();

  // ---- Normalize + write out, fully coalesced (256 threads = one row per pass)
#pragma unroll 2
  for (int rr = 0; rr < 16; ++rr) {
    const float v = ldsS[rr * SS + tid];
    out[(rowbase + rr) * KCLS + tid] = expf(v - rowMax[rr]) * rowInv[rr];
  }
}

// ---------------------------------------------------------------------------
// Launch: inputs are (x, centers, Droot) fp32; output resp fp32 [4096,256].
// Workspace: W (256*512 f32) + bias (256 f32) = 525,312 bytes.
// ---------------------------------------------------------------------------
extern "C" void kernel_launch(void* const* d_in, const int* in_sizes, int n_in,
                              void* d_out, int out_size, void* d_ws, size_t ws_size,
                              hipStream_t stream) {
  const float* x       = (const float*)d_in[0];
  const float* centers = (const float*)d_in[1];
  const float* Droot   = (const float*)d_in[2];
  float* W    = (float*)d_ws;
  float* bias = W + (size_t)KCLS * AD;
  float* out  = (float*)d_out;

  gmm_prep_kernel<<<KCLS, 256, 0, stream>>>(centers, Droot, W, bias);
  gmm_main_kernel<<<BATCH / 16, 256, 0, stream>>>(x, W, bias, out);
}